// ChamferDistance_32195074851184
// MI455X (gfx1250) — compile-verified
//
#include <hip/hip_runtime.h>

typedef __attribute__((ext_vector_type(2))) float v2f;
typedef __attribute__((ext_vector_type(8))) float v8f;

constexpr int NPTS  = 8192;
constexpr int NB    = 4;
constexpr int TILES = NPTS / 16;   // 512 column tiles

// Single-VALU-op float min: avoids fminf's NaN-canonicalize ops and the
// v_cmp/v_cndmask scalar-register serialization of compare-select.
__device__ __forceinline__ void vmin(float& t, float d) {
    asm("v_min_num_f32 %0, %0, %1" : "+v"(t) : "v"(d));
}

// One wave per (dir, batch, 16-row tile). Computes, for each of its 16 P-points,
// min over all 8192 Q-points of ||p - q||, using V_WMMA_F32_16X16X4_F32 for the
// -2*p.q cross term (K=4, d=3 zero-padded), with ||q||^2 folded into the C
// accumulator and ||p||^2 added after the min reduction.
__global__ __launch_bounds__(32) void chamfer_min_kernel(
    const float* __restrict__ x, const float* __restrict__ y,
    float* __restrict__ minvals)
{
    const int rowTile = blockIdx.x;
    const int b       = blockIdx.y;
    const int dir     = blockIdx.z;
    const float* __restrict__ P = (dir == 0) ? x : y;
    const float* __restrict__ Q = (dir == 0) ? y : x;

    const int lane = threadIdx.x;     // 0..31, wave32
    const int half = lane >> 4;       // 0 -> K={0,1}, 1 -> K={2,3(pad)}
    const int l15  = lane & 15;

    const size_t base = (size_t)b * NPTS * 3;
    const int m = rowTile * 16 + l15;

    // Load this lane's row point fully; build A tile (16x4 f32) branch-free.
    // a[v] = -2 * P[m][2*half + v], K==3 zero-padded.
    v2f a;
    float ppl;
    {
        const float* __restrict__ p = P + base + (size_t)m * 3;
        float p0 = p[0], p1 = p[1], p2 = p[2];
        ppl = p0 * p0 + p1 * p1 + p2 * p2;
        a.x = -2.0f * (half ? p2 : p0);
        a.y = half ? 0.0f : (-2.0f * p1);
    }

    float t[8];
#pragma unroll
    for (int v = 0; v < 8; ++v) t[v] = 3.0e38f;

#pragma unroll 4
    for (int ct = 0; ct < TILES; ++ct) {
        const int n = ct * 16 + l15;
        // Load full column point: gives both the B elements and ||q||^2 locally
        // (no cross-lane exchange, no divergent guard in the hot loop).
        const float* __restrict__ q = Q + base + (size_t)n * 3;
        float c0 = q[0], c1 = q[1], c2 = q[2];
        float qq = c0 * c0 + c1 * c1 + c2 * c2;

        v2f bm;                        // B tile (4x16): bm[v] = Q[n][2*half + v]
        bm.x = half ? c2 : c0;
        bm.y = half ? 0.0f : c1;

        v8f c = {qq, qq, qq, qq, qq, qq, qq, qq};

        // D[v][lane] = ||q_n||^2 - 2 * p_{rowBase+v+8*half} . q_n
        v8f d = __builtin_amdgcn_wmma_f32_16x16x4_f32(
            /*neg_a=*/false, a, /*neg_b=*/false, bm,
            /*c_mod=*/(short)0, c, /*reuse_a=*/false, /*reuse_b=*/false);

#pragma unroll
        for (int v = 0; v < 8; ++v) vmin(t[v], d[v]);
    }

    // Min across the 16 lanes of each half (columns of the tile).
#pragma unroll
    for (int mask = 1; mask < 16; mask <<= 1) {
#pragma unroll
        for (int v = 0; v < 8; ++v) vmin(t[v], __shfl_xor(t[v], mask, 32));
    }

    float* __restrict__ outd = minvals + ((size_t)dir * NB + b) * NPTS;
#pragma unroll
    for (int v = 0; v < 8; ++v) {
        // row handled by slot v in this half: rowTile*16 + v + 8*half;
        // its ||p||^2 lives in lane (v + 8*half) (uniform shuffle, all lanes exec)
        float ppr = __shfl(ppl, v + 8 * half, 32);
        float val = t[v] + ppr;
        if (l15 == 0) {
            int row = rowTile * 16 + v + 8 * half;
            outd[row] = sqrtf(fmaxf(val, 0.0f));
        }
    }
}

// Final reduction: result = sum(all min distances) / (n * bs), since the
// reference's per-batch means + batch mean collapse into one global mean
// (both directions share the same 1/(n*bs) factor and are summed).
__global__ void chamfer_reduce_kernel(const float* __restrict__ minvals,
                                      float* __restrict__ out)
{
    __shared__ float s[256];
    const int tid = threadIdx.x;
    float acc = 0.0f;
    for (int i = tid; i < 2 * NB * NPTS; i += 256) acc += minvals[i];
    s[tid] = acc;
    __syncthreads();
    for (int stride = 128; stride > 0; stride >>= 1) {
        if (tid < stride) s[tid] += s[tid + stride];
        __syncthreads();
    }
    if (tid == 0) out[0] = s[0] * (1.0f / (float)(NB * NPTS));
}

extern "C" void kernel_launch(void* const* d_in, const int* in_sizes, int n_in,
                              void* d_out, int out_size, void* d_ws, size_t ws_size,
                              hipStream_t stream)
{
    const float* x = (const float*)d_in[0];
    const float* y = (const float*)d_in[1];
    float* minvals = (float*)d_ws;        // 2 * 4 * 8192 floats = 256 KB
    float* out     = (float*)d_out;

    dim3 grid(TILES, NB, 2);              // (rowTile, batch, direction)
    chamfer_min_kernel<<<grid, 32, 0, stream>>>(x, y, minvals);
    chamfer_reduce_kernel<<<1, 256, 0, stream>>>(minvals, out);
}